// FeatureAlign_64063732187249
// MI455X (gfx1250) — compile-verified
//
#include <hip/hip_runtime.h>
#include <hip/hip_bf16.h>
#include <math.h>

typedef __attribute__((ext_vector_type(2))) float v2f;
typedef __attribute__((ext_vector_type(8))) float v8f;

#define Nn 4
#define Dd 256
#define Hd 64
#define Wd 64
#define HW (Hd * Wd)

// ---------------------------------------------------------------------------
// Prep kernel: per-pixel L2 norms of cur and prev over the D=256 channels.
// cnF/pnF live in d_ws (2 * N*HW floats = 128 KB). Coalesced along W.
// ---------------------------------------------------------------------------
__global__ __launch_bounds__(256) void FeatureAlign_norms(
    const float* __restrict__ cur, const float* __restrict__ prev,
    float* __restrict__ cnF, float* __restrict__ pnF) {
  const int idx = blockIdx.x * 256 + threadIdx.x;   // over N*HW = 16384
  const int n = idx >> 12;
  const int pix = idx & (HW - 1);
  const size_t base = (size_t)n * Dd * HW + pix;
  float cs = 0.0f, ps = 0.0f;
  for (int d = 0; d < Dd; ++d) {
    const float c = cur[base + (size_t)d * HW];
    const float p = prev[base + (size_t)d * HW];
    cs += c * c;
    ps += p * p;
  }
  cnF[idx] = sqrtf(cs) + 1e-8f;
  pnF[idx] = sqrtf(ps) + 1e-8f;
}

// ---------------------------------------------------------------------------
// Main kernel. One block = one (n, y) row. 128 threads = 4 waves; wave w owns
// x-tile [16w, 16w+16).
// Phase 1: per wave, 8 WMMA f32 16x16x4 accumulations over K=256:
//   A = cur pixels x0..x0+15 (rows), B = prev row (y+dy) pixels x0-1+16h+j,
//   dy in {-1,0,1,2}, h in {0,1}. Entry (i,j) = dot(cur[x0+i], prev[col]),
//   dx = 16h + j - i - 1, covering all needed dx in {-1..2}. OOB neighbor
//   coordinates are CLAMPED (loads always valid); consumers of OOB entries
//   are masked in phase 2. Addressing: uniform SGPR channel base + invariant
//   unsigned 32-bit lane offsets (GVS form); fragments double-buffered to
//   overlap VMEM latency with the WMMA pipe.
// Phase 2: normalize -> affinity band, masked 3x3 mass, bound cond, and the
//   4 {-1,2}^2 aggregation weights (reference's "tuple" offsets).
// Phase 3: out[d,y,x] = cond ? mem + (d>0 ? sum_off w*mem[neighbor] : 0) : 0.
// ---------------------------------------------------------------------------
__global__ __launch_bounds__(128) void FeatureAlign_kernel(
    const float* __restrict__ cur, const float* __restrict__ prev,
    const float* __restrict__ mem, const float* __restrict__ cnF,
    const float* __restrict__ pnF, float* __restrict__ out) {
  __shared__ float ldsD[4][4][16][32];   // [wave][dyIdx][i][j] raw dots
  __shared__ float ldsW[4][Wd];          // 4 agg weights per pixel
  __shared__ int   ldsC[Wd];             // mass-bound condition

  const int nb = blockIdx.x;        // n*H + y
  const int n = nb >> 6;
  const int y = nb & (Hd - 1);
  const int t = threadIdx.x;
  const int wave = t >> 5;
  const int lane = t & 31;
  const int g = lane >> 4;          // lane group selects K pair
  const int li = lane & 15;         // M (A) / N (B) index
  const int x0 = wave << 4;

  const float* curN = cur + (size_t)n * Dd * HW;
  const float* prevN = prev + (size_t)n * Dd * HW;
  const float* memN = mem + (size_t)n * Dd * HW;
  float* outN = out + (size_t)n * Dd * HW;

  // Loop-invariant unsigned lane offsets (channel-pair select 2g folded in, so
  // the per-iteration base pointer stays uniform / SGPR-resident and the loads
  // can use saddr + 32-bit vaddr).
  const unsigned voffA = (unsigned)(2 * g * HW + y * Wd + x0 + li);
  unsigned voffB[4][2];
#pragma unroll
  for (int dyi = 0; dyi < 4; ++dyi) {
    const int dy = (dyi == 3) ? 2 : (dyi - 1);    // -1, 0, 1, 2
    int yy = y + dy;
    yy = yy < 0 ? 0 : (yy > Hd - 1 ? Hd - 1 : yy);
#pragma unroll
    for (int h = 0; h < 2; ++h) {
      int bx = x0 - 1 + 16 * h + li;
      bx = bx < 0 ? 0 : (bx > Wd - 1 ? Wd - 1 : bx);
      voffB[dyi][h] = (unsigned)(2 * g * HW + yy * Wd + bx);
    }
  }
  __builtin_assume(voffA < 3u * HW);

  v8f acc[4][2];
#pragma unroll
  for (int dyi = 0; dyi < 4; ++dyi)
#pragma unroll
    for (int h = 0; h < 2; ++h) acc[dyi][h] = {};

  auto loadA = [&](int k, v2f& a) {
    const float* pk = curN + (size_t)(unsigned)k * HW;    // uniform base
    a.x = pk[voffA];
    a.y = pk[voffA + HW];
  };
  auto loadB = [&](int k, v2f (&b)[4][2]) {
    const float* pk = prevN + (size_t)(unsigned)k * HW;   // uniform base
#pragma unroll
    for (int dyi = 0; dyi < 4; ++dyi)
#pragma unroll
      for (int h = 0; h < 2; ++h) {
        b[dyi][h].x = pk[voffB[dyi][h]];
        b[dyi][h].y = pk[voffB[dyi][h] + HW];
      }
  };
  auto domma = [&](const v2f& a, const v2f (&b)[4][2]) {
#pragma unroll
    for (int dyi = 0; dyi < 4; ++dyi)
#pragma unroll
      for (int h = 0; h < 2; ++h)
        acc[dyi][h] = __builtin_amdgcn_wmma_f32_16x16x4_f32(
            false, a, false, b[dyi][h], (short)0, acc[dyi][h], false, false);
  };

  // Software-pipelined K loop: prefetch chunk k+4 while multiplying chunk k.
  v2f a0, a1;
  v2f b0[4][2], b1[4][2];
  loadA(0, a0);
  loadB(0, b0);
  for (int k = 0; k < Dd; k += 8) {
    loadA(k + 4, a1);
    loadB(k + 4, b1);
    domma(a0, b0);
    const int kq = (k + 8 < Dd) ? (k + 8) : 0;    // tail prefetch stays valid
    loadA(kq, a0);
    loadB(kq, b0);
    domma(a1, b1);
  }

  // Spill D tiles: VGPR r holds row M = r + 8g, column N = li.
#pragma unroll
  for (int dyi = 0; dyi < 4; ++dyi)
#pragma unroll
    for (int h = 0; h < 2; ++h)
#pragma unroll
      for (int r = 0; r < 8; ++r)
        ldsD[wave][dyi][r + 8 * g][16 * h + li] = acc[dyi][h][r];

  __syncthreads();

  // Phase 2: one thread per pixel computes mass, cond, and the 4 weights.
  const int nb2 = n * HW;
  if (t < Wd) {
    const int x = t, w = x >> 4, i = x & 15;
    const float cn = cnF[nb2 + y * Wd + x];
    float mass = 0.0f;
#pragma unroll
    for (int dyi = 0; dyi < 3; ++dyi) {          // dy = -1,0,1
      const int yy = y + dyi - 1;
      const bool rowv = (yy >= 0) && (yy < Hd);
#pragma unroll
      for (int dx = -1; dx <= 1; ++dx) {
        const int px = x + dx;
        if (rowv && px >= 0 && px < Wd) {
          const float pn = pnF[nb2 + yy * Wd + px];
          const float a = ldsD[w][dyi][i][i + 1 + dx] / (cn * pn);
          mass += fmaxf(a, 0.0f);
        }
      }
    }
    ldsC[x] = (mass > -10.0f) && (mass < 10.0f);
    const float msafe = (mass > 0.0f) ? mass : 1.0f;
#pragma unroll
    for (int oy = 0; oy < 2; ++oy) {
      const int dy = oy ? 2 : -1;
      const int dyi = oy ? 3 : 0;
      const int py = y + dy;
#pragma unroll
      for (int ox = 0; ox < 2; ++ox) {
        const int dx = ox ? 2 : -1;
        const int px = x + dx;
        float wv = 0.0f;
        if (py >= 0 && py < Hd && px >= 0 && px < Wd) {
          const float pn = pnF[nb2 + py * Wd + px];
          const float a = ldsD[w][dyi][i][i + 1 + dx] / (cn * pn);
          wv = (a > 0.0f) ? (a / msafe) : 0.0f;
        }
        ldsW[oy * 2 + ox][x] = wv;
      }
    }
  }
  __syncthreads();

  // Phase 3: stream mem -> out for all 256 channels of this row.
  const int yb = y * Wd;
  for (int it = 0; it < (Dd * Wd) / 128; ++it) {
    const int idx2 = t + 128 * it;
    const int d = idx2 >> 6;
    const int x = idx2 & (Wd - 1);
    const size_t base = (size_t)d * HW;
    const float m0 = memN[base + yb + x];
    float agg = 0.0f;
    if (d > 0) {
#pragma unroll
      for (int off = 0; off < 4; ++off) {
        const int dy = (off < 2) ? -1 : 2;
        const int dx = (off & 1) ? 2 : -1;
        int py = y + dy; py = py < 0 ? 0 : (py > Hd - 1 ? Hd - 1 : py);
        int px = x + dx; px = px < 0 ? 0 : (px > Wd - 1 ? Wd - 1 : px);
        agg += ldsW[off][x] * memN[base + py * Wd + px];
      }
    }
    outN[base + yb + x] = ldsC[x] ? (m0 + agg) : 0.0f;
  }
}

extern "C" void kernel_launch(void* const* d_in, const int* in_sizes, int n_in,
                              void* d_out, int out_size, void* d_ws, size_t ws_size,
                              hipStream_t stream) {
  const float* cur = (const float*)d_in[0];
  const float* prev = (const float*)d_in[1];
  const float* mem = (const float*)d_in[2];
  float* out = (float*)d_out;
  (void)in_sizes; (void)n_in; (void)out_size; (void)ws_size;

  float* cnF = (float*)d_ws;              // N*HW floats
  float* pnF = cnF + (size_t)Nn * HW;     // N*HW floats  (128 KB total)

  FeatureAlign_norms<<<dim3((Nn * HW) / 256), dim3(256), 0, stream>>>(
      cur, prev, cnF, pnF);
  FeatureAlign_kernel<<<dim3(Nn * Hd), dim3(128), 0, stream>>>(
      cur, prev, mem, cnF, pnF, out);
}